// NetworkBasic_52982716563739
// MI455X (gfx1250) — compile-verified
//
#include <hip/hip_runtime.h>

// SLAYER SRM forward on MI455X (gfx1250), wave32.
// Convs mapped to V_WMMA_F32_16X16X4_F32 (K=4 == C_in), shift-packed so
// N = 4 output channels x 4 output-x shifts (full 16-wide N usage per strip).
// Temporal recurrences (alpha-PSP + refractory spike dynamics) live in VGPRs.

typedef __attribute__((ext_vector_type(2))) float v2f;
typedef __attribute__((ext_vector_type(8))) float v8f;

#define T_STEPS 50
#define NB 8
#define CC 4
#define HH 128
#define WW 128
// zero-padded intermediate layouts: [N, T, Hp, Wp, C] (C fastest -> one b64
// load fetches a lane's ci-pair of the WMMA A matrix)
#define HP1 132   // conv1 halo 2 each side
#define WP1 132
#define HP2 130   // conv2 halo 1 each side
#define WP2 130

// exact reference constants
__device__ __constant__ float kRA  = 0.36787944117144233f;  // e^-1      (psp1 r, spike1 refr r)
__device__ __constant__ float kCA  = 2.71828182845904523f;  // e         (psp1 coef, tau=1)
__device__ __constant__ float kCF1 = -54.3656365691809047f; // -20e      (spike1 refr coef)
__device__ __constant__ float kRB  = 0.60653065971263342f;  // e^-0.5    (psp2 r, spike2 refr r)
__device__ __constant__ float kCB  = 1.35914091422952262f;  // e/2       (psp2 coef, tau=2)
__device__ __constant__ float kCF2 = -54.3656365691809047f; // -40e/2    (spike2 refr coef)

// ---------------------------------------------------------------------------
// Zero the padded scratch buffers (borders must read as 0 every call; the
// harness poisons d_ws once and never re-poisons).
__global__ void slayer_zero_ws(float4* __restrict__ p, long n4) {
  long i = (long)blockIdx.x * blockDim.x + threadIdx.x;
  long stride = (long)gridDim.x * blockDim.x;
  float4 z = make_float4(0.f, 0.f, 0.f, 0.f);
  for (; i < n4; i += stride) p[i] = z;
}

// ---------------------------------------------------------------------------
// Pass 1: psp1 (tau=1) recurrence per neuron, transposing [N,C,H,W,T] ->
// padded [N,T,HP1,WP1,C]. Writes are coalesced (c,w fastest across lanes).
__global__ void slayer_psp1(const float* __restrict__ x, float* __restrict__ p1) {
  int gid = blockIdx.x * blockDim.x + threadIdx.x;           // 524288 threads
  int c = gid & 3;
  int w = (gid >> 2) & 127;
  int h = (gid >> 9) & 127;
  int n = gid >> 16;
  const float* xi = x + ((((long)n * CC + c) * HH + h) * WW + w) * T_STEPS;
  float s1 = 0.f, s2 = 0.f;
  long o = ((((long)n * T_STEPS) * HP1 + (h + 2)) * WP1 + (w + 2)) * CC + c;
  const long ostride = (long)HP1 * WP1 * CC;
  for (int t = 0; t < T_STEPS; ++t, o += ostride) {
    float out = kCA * s2;          // kernel value 0 at lag 0
    s1 = kRA * s1 + xi[t];
    s2 = kRA * s2 + kRA * s1;
    p1[o] = out;
  }
}

// ---------------------------------------------------------------------------
// Pass 2: conv1 (5x5, 4->4) via 40 chained f32 WMMAs per row-strip per t,
// fused with spike1 dynamics (theta=20, tauRef=1) and psp2 (tau=2).
// Block = 8 waves; wave wv owns output row y = blockIdx.x*8+wv, loops t with
// all recurrence states in registers. Output: padded psp2-out buffer.
__global__ void __launch_bounds__(256) slayer_layer1(const float* __restrict__ p1,
                                                     const float* __restrict__ w1,
                                                     float* __restrict__ p2) {
  // Precompute the 40 shift-packed B matrices (kh in 0..4, j in 0..7):
  // B[ci, co*4+s] = w1[co,ci,kh,j-s] if 0<=j-s<5 else 0.
  // Layout matches WMMA B VGPR striping: Bm[pair*64 + lane*2 + kslot].
  __shared__ float Bm[40 * 64];
  int tid = threadIdx.x;
  for (int idx = tid; idx < 40 * 64; idx += 256) {
    int p = idx >> 6, within = idx & 63;
    int lane = within >> 1, ks = within & 1;
    int kh = p >> 3, j = p & 7;
    int K  = ks + ((lane & 16) ? 2 : 0);      // ci
    int Nn = lane & 15, co = Nn >> 2, s = Nn & 3;
    int kw = j - s;
    Bm[idx] = (kw >= 0 && kw < 5) ? w1[((co * CC + K) * 5 + kh) * 5 + kw] : 0.0f;
  }
  __syncthreads();

  int lane = tid & 31, wv = tid >> 5;
  int n = blockIdx.z, xs = blockIdx.y, ys = blockIdx.x;
  int y  = ys * 8 + wv;                 // output row 0..127
  int x0 = xs * 64;                     // 64-pixel x strip
  int Mi = lane & 15;                   // A-matrix M index (base pixel)
  int hi = lane >> 4;                   // 0: K=0,1  1: K=2,3
  int ci0 = hi * 2;
  int Nn = lane & 15, co = Nn >> 2, s = Nn & 3;

  float rs1[8], rs2[8], q1[8], q2[8];   // spike1 refractory + psp2 states
#pragma unroll
  for (int r = 0; r < 8; ++r) { rs1[r] = rs2[r] = q1[r] = q2[r] = 0.f; }

  const float* arow0 = p1 + ((((long)n * T_STEPS) * HP1 + y) * WP1 + (x0 + 4 * Mi)) * CC + ci0;
  const long sliceA = (long)HP1 * WP1 * CC;
  float* orow0 = p2 + (((long)n * T_STEPS) * HP2 + (y + 1)) * WP2 * CC;
  const long sliceO = (long)HP2 * WP2 * CC;
  int xb = x0 + 32 * hi + s;            // output x = xb + 4*r

  for (int t = 0; t < T_STEPS; ++t) {
    v8f acc = {0.f, 0.f, 0.f, 0.f, 0.f, 0.f, 0.f, 0.f};
    const float* ar = arow0 + (long)t * sliceA;
#pragma unroll
    for (int kh = 0; kh < 5; ++kh) {
#pragma unroll
      for (int j = 0; j < 8; ++j) {
        v2f a = *(const v2f*)(ar + (long)kh * (WP1 * CC) + j * CC);    // global_load_b64
        v2f b = *(const v2f*)(&Bm[(kh * 8 + j) * 64 + lane * 2]);      // ds_load_b64
        acc = __builtin_amdgcn_wmma_f32_16x16x4_f32(false, a, false, b,
                                                    (short)0, acc, false, false);
      }
    }
    // spike1 (theta=20) + psp2 per element; acc[r] = u1 at (co, x = xb+4r)
    float* orow = orow0 + (long)t * sliceO;
#pragma unroll
    for (int r = 0; r < 8; ++r) {
      float m   = acc[r] + kCF1 * rs2[r];            // refr kernel[0]=0
      float spk = (m >= 20.0f) ? 1.0f : 0.0f;
      rs1[r] = kRA * rs1[r] + spk;
      rs2[r] = kRA * rs2[r] + kRA * rs1[r];
      float out2 = kCB * q2[r];                      // psp2 out uses pre-update q2
      q1[r] = kRB * q1[r] + spk;
      q2[r] = kRB * q2[r] + kRB * q1[r];
      orow[(xb + 4 * r + 1) * CC + co] = out2;       // padded x index = x+1
    }
  }
}

// ---------------------------------------------------------------------------
// Pass 3: conv2 (3x3, 4->4) via 18 chained WMMAs per strip per t, fused with
// spike2 dynamics (theta=40, tauRef=2). Writes s2 in reference [N,C,H,W,T].
__global__ void __launch_bounds__(256) slayer_layer2(const float* __restrict__ p2,
                                                     const float* __restrict__ w2,
                                                     float* __restrict__ out) {
  __shared__ float Bm[18 * 64];
  int tid = threadIdx.x;
  for (int idx = tid; idx < 18 * 64; idx += 256) {
    int p = idx >> 6, within = idx & 63;
    int lane = within >> 1, ks = within & 1;
    int kh = p / 6, j = p % 6;
    int K  = ks + ((lane & 16) ? 2 : 0);
    int Nn = lane & 15, co = Nn >> 2, s = Nn & 3;
    int kw = j - s;
    Bm[idx] = (kw >= 0 && kw < 3) ? w2[((co * CC + K) * 3 + kh) * 3 + kw] : 0.0f;
  }
  __syncthreads();

  int lane = tid & 31, wv = tid >> 5;
  int n = blockIdx.z, xs = blockIdx.y, ys = blockIdx.x;
  int y  = ys * 8 + wv;
  int x0 = xs * 64;
  int Mi = lane & 15;
  int hi = lane >> 4;
  int ci0 = hi * 2;
  int Nn = lane & 15, co = Nn >> 2, s = Nn & 3;

  float rs1[8], rs2[8];
#pragma unroll
  for (int r = 0; r < 8; ++r) { rs1[r] = rs2[r] = 0.f; }

  const float* arow0 = p2 + ((((long)n * T_STEPS) * HP2 + y) * WP2 + (x0 + 4 * Mi)) * CC + ci0;
  const long sliceA = (long)HP2 * WP2 * CC;
  int xb = x0 + 32 * hi + s;
  // out[n][co][y][x][t]
  long obase = ((((long)n * CC + co) * HH + y) * WW + xb) * T_STEPS;

  for (int t = 0; t < T_STEPS; ++t) {
    v8f acc = {0.f, 0.f, 0.f, 0.f, 0.f, 0.f, 0.f, 0.f};
    const float* ar = arow0 + (long)t * sliceA;
#pragma unroll
    for (int kh = 0; kh < 3; ++kh) {
#pragma unroll
      for (int j = 0; j < 6; ++j) {
        v2f a = *(const v2f*)(ar + (long)kh * (WP2 * CC) + j * CC);
        v2f b = *(const v2f*)(&Bm[(kh * 6 + j) * 64 + lane * 2]);
        acc = __builtin_amdgcn_wmma_f32_16x16x4_f32(false, a, false, b,
                                                    (short)0, acc, false, false);
      }
    }
#pragma unroll
    for (int r = 0; r < 8; ++r) {
      float m   = acc[r] + kCF2 * rs2[r];
      float spk = (m >= 40.0f) ? 1.0f : 0.0f;
      rs1[r] = kRB * rs1[r] + spk;
      rs2[r] = kRB * rs2[r] + kRB * rs1[r];
      out[obase + (long)(4 * r) * T_STEPS + t] = spk;
    }
  }
}

// ---------------------------------------------------------------------------
extern "C" void kernel_launch(void* const* d_in, const int* in_sizes, int n_in,
                              void* d_out, int out_size, void* d_ws, size_t ws_size,
                              hipStream_t stream) {
  (void)in_sizes; (void)n_in; (void)out_size; (void)ws_size;
  const float* x  = (const float*)d_in[0];   // [8,4,128,128,50]
  const float* w1 = (const float*)d_in[1];   // [4,4,5,5]
  const float* w2 = (const float*)d_in[2];   // [4,4,3,3]
  // d_in[3] (w3) feeds only the dead third layer -> unused.

  // scratch: p1 = [N,T,132,132,4] then p2 = [N,T,130,130,4]; total ~220 MB.
  float* p1 = (float*)d_ws;
  const long p1_elems = (long)NB * T_STEPS * HP1 * WP1 * CC;   // 27,878,400
  float* p2 = p1 + p1_elems;
  const long p2_elems = (long)NB * T_STEPS * HP2 * WP2 * CC;   // 27,040,000
  const long total4 = (p1_elems + p2_elems) / 4;

  slayer_zero_ws<<<2048, 256, 0, stream>>>((float4*)d_ws, total4);
  slayer_psp1<<<2048, 256, 0, stream>>>(x, p1);                // 524288 neurons
  dim3 grid(16, 2, NB);                                        // (ystrip, xstrip, n)
  slayer_layer1<<<grid, 256, 0, stream>>>(p1, w1, p2);
  slayer_layer2<<<grid, 256, 0, stream>>>(p2, w2, (float*)d_out);
}